// Attention_8546984919065
// MI455X (gfx1250) — compile-verified
//
#include <hip/hip_runtime.h>
#include <hip/hip_bf16.h>
#include <math.h>

// ---------------- problem dims ----------------
#define B_DIM   64
#define S_DIM   4096
#define H_DIM   512
#define CHUNKS  8
#define CS      (S_DIM / CHUNKS)   // 512 seq positions per chunk
#define TS      32                 // tile: 32 seq rows staged in LDS
#define NT      (CS / TS)          // 16 tiles per chunk
#define TSTRIDE 520                // padded row stride (dwords): keeps half-wave
                                   // B-operand reads on disjoint LDS banks
#define NWAVES  4
#define BLK     (NWAVES * 32)
#define OPS_PER_TILE (TS * (H_DIM / 4) / BLK)   // 32 async b128 ops per thread

typedef float v2f __attribute__((ext_vector_type(2)));
typedef float v8f __attribute__((ext_vector_type(8)));
typedef int   v4i __attribute__((ext_vector_type(4)));

typedef __attribute__((address_space(1))) v4i gas_v4i;   // global (device) AS
typedef __attribute__((address_space(3))) v4i las_v4i;   // LDS (shared) AS

#if defined(__has_builtin)
#if __has_builtin(__builtin_amdgcn_global_load_async_to_lds_b128) && \
    __has_builtin(__builtin_amdgcn_s_wait_asynccnt)
#define USE_ASYNC_LDS 1
#else
#define USE_ASYNC_LDS 0
#endif
#else
#define USE_ASYNC_LDS 0
#endif

__device__ __forceinline__ float wave_sum32(float v) {
#pragma unroll
  for (int off = 16; off >= 1; off >>= 1) v += __shfl_xor(v, off, 32);
  return v;
}
__device__ __forceinline__ float wave_max32(float v) {
#pragma unroll
  for (int off = 16; off >= 1; off >>= 1) v = fmaxf(v, __shfl_xor(v, off, 32));
  return v;
}

// Issue the 32 per-thread async b128 copies staging one 32x512 fp32 tile.
__device__ __forceinline__ void issue_tile(float* lds_buf, const float* gt, int tid) {
#if USE_ASYNC_LDS
  for (int idx = tid; idx < TS * (H_DIM / 4); idx += BLK) {
    const int r  = idx >> 7;              // row 0..31
    const int c4 = idx & 127;             // float4 within row
    const float* src = gt + (size_t)r * H_DIM + c4 * 4;
    float* dst = &lds_buf[r * TSTRIDE + c4 * 4];
    __builtin_amdgcn_global_load_async_to_lds_b128(
        (gas_v4i*)(void*)src, (las_v4i*)(void*)dst, 0, 0);
  }
#else
  for (int idx = tid; idx < TS * (H_DIM / 4); idx += BLK) {
    const int r  = idx >> 7;
    const int c4 = idx & 127;
    *(float4*)&lds_buf[r * TSTRIDE + c4 * 4] =
        *(const float4*)(gt + (size_t)r * H_DIM + c4 * 4);
  }
#endif
}

// ---------------------------------------------------------------------------
// Pass 1: one workgroup per (chunk, batch). Streams X exactly once via
// double-buffered async global->LDS staging. Online softmax with WMMA
// f32 16x16x4 accumulation of the pooled partial.
// ---------------------------------------------------------------------------
__global__ __launch_bounds__(BLK) void attn_pass1(
    const float* __restrict__ x,       // [B, S, H]
    const float* __restrict__ w,       // [H]
    float* __restrict__ scores_out,    // [B, S]  (raw scores, d_out weights region)
    float* __restrict__ partial,       // [B, CHUNKS, H]
    float* __restrict__ ml)            // [B, CHUNKS, 2]
{
  __shared__ float tile[2][TS * TSTRIDE];
  __shared__ float sscore[TS];

  const int b     = blockIdx.y;
  const int chunk = blockIdx.x;
  const int tid   = threadIdx.x;
  const int lane  = tid & 31;
  const int wv    = tid >> 5;
  const int h0w   = wv * 128;               // this wave's 128-wide h slice

  const float* xb = x + ((size_t)b * S_DIM) * (size_t)H_DIM;
  const int s_base = chunk * CS;

  // w slices for the score dot: lane covers h = lane + 32*j
  float wreg[16];
#pragma unroll
  for (int j = 0; j < 16; ++j) wreg[j] = w[lane + 32 * j];

  float m_run = -1e30f;
  float l_run = 0.0f;
  v8f   acc[8] = {};                        // 8 WMMA C tiles (h0w .. h0w+127)

  const int halfsel = (lane >> 4) << 1;     // 0 for lanes 0-15, 2 for 16-31
  const int col     = lane & 15;

  // prologue: stage tile 0 into buffer 0
  issue_tile(tile[0], xb + (size_t)s_base * H_DIM, tid);

  for (int t = 0; t < NT; ++t) {
    const int srow = s_base + t * TS;
    const float* buf = tile[t & 1];

    // issue next tile into the other buffer, then wait only for this tile
    if (t + 1 < NT) {
      issue_tile((float*)tile[(t + 1) & 1],
                 xb + (size_t)(srow + TS) * H_DIM, tid);
#if USE_ASYNC_LDS
      __builtin_amdgcn_s_wait_asynccnt(OPS_PER_TILE);  // tile t landed (in-order)
#endif
    } else {
#if USE_ASYNC_LDS
      __builtin_amdgcn_s_wait_asynccnt(0);
#endif
    }
    __syncthreads();

    // ---- scores: wave wv computes s_local = wv*8 + i ----
#pragma unroll
    for (int i = 0; i < 8; ++i) {
      const int sl = wv * 8 + i;
      const float* row = &buf[sl * TSTRIDE];
      float p = 0.0f;
#pragma unroll
      for (int j = 0; j < 16; ++j) p = fmaf(row[lane + 32 * j], wreg[j], p);
      p = wave_sum32(p);
      if (lane == 0) {
        sscore[sl] = p;
        scores_out[(size_t)b * S_DIM + srow + sl] = p;   // raw score stash
      }
    }
    __syncthreads();

    // ---- online softmax state update (redundant per wave, identical) ----
    const float sc_l = sscore[lane];            // lane l holds score of s_local=l
    const float mt   = wave_max32(sc_l);
    const float newm = fmaxf(m_run, mt);
    const float scale = __expf(m_run - newm);
    const float p_l   = __expf(sc_l - newm);
    const float tsum  = wave_sum32(p_l);
    l_run = l_run * scale + tsum;
    m_run = newm;
#pragma unroll
    for (int tt = 0; tt < 8; ++tt) acc[tt] *= scale;

    // ---- pooled accumulation: D = A(p replicated) x B(x tile) + C ----
#pragma unroll
    for (int kb = 0; kb < 8; ++kb) {
      const int rb = kb * 4 + halfsel;          // A/B half-wave row base
      v2f A;
      A.x = __shfl(p_l, rb, 32);                // K rows rb, rb+1 of p
      A.y = __shfl(p_l, rb + 1, 32);
#pragma unroll
      for (int tt = 0; tt < 8; ++tt) {
        const int hc = h0w + tt * 16 + col;
        v2f Bm;
        Bm.x = buf[rb * TSTRIDE + hc];
        Bm.y = buf[(rb + 1) * TSTRIDE + hc];
        acc[tt] = __builtin_amdgcn_wmma_f32_16x16x4_f32(
            false, A, false, Bm, (short)0, acc[tt], false, false);
      }
    }
    __syncthreads();   // all waves done with buf before it is re-staged
  }

  // ---- emit chunk partials ----
  if (tid == 0) {
    ml[((size_t)b * CHUNKS + chunk) * 2 + 0] = m_run;
    ml[((size_t)b * CHUNKS + chunk) * 2 + 1] = l_run;
  }
  if (lane < 16) {   // D rows are replicated; row M=0 lives in C[0], lanes 0-15
#pragma unroll
    for (int tt = 0; tt < 8; ++tt) {
      partial[((size_t)b * CHUNKS + chunk) * H_DIM + h0w + tt * 16 + lane] =
          acc[tt][0];
    }
  }
}

// ---------------------------------------------------------------------------
// Pass 2: per-batch combine of the 8 chunk partials; normalize weights
// in-place from stashed raw scores. Tiny (~2 MB traffic).
// ---------------------------------------------------------------------------
__global__ __launch_bounds__(128) void attn_pass2(
    float* __restrict__ pooled,        // d_out [B, H]
    float* __restrict__ weights,       // d_out + B*H, holds raw scores on entry
    const float* __restrict__ partial, // [B, CHUNKS, H]
    const float* __restrict__ ml)      // [B, CHUNKS, 2]
{
  const int b = blockIdx.x;
  const int tid = threadIdx.x;

  float mc[CHUNKS], lc[CHUNKS];
  float mg = -1e30f;
#pragma unroll
  for (int c = 0; c < CHUNKS; ++c) {
    mc[c] = ml[((size_t)b * CHUNKS + c) * 2 + 0];
    lc[c] = ml[((size_t)b * CHUNKS + c) * 2 + 1];
    mg = fmaxf(mg, mc[c]);
  }
  float L = 0.0f;
  float ec[CHUNKS];
#pragma unroll
  for (int c = 0; c < CHUNKS; ++c) { ec[c] = __expf(mc[c] - mg); L += lc[c] * ec[c]; }
  const float invL = 1.0f / L;

  for (int h = tid; h < H_DIM; h += 128) {
    float a = 0.0f;
#pragma unroll
    for (int c = 0; c < CHUNKS; ++c)
      a += partial[((size_t)b * CHUNKS + c) * H_DIM + h] * ec[c];
    pooled[(size_t)b * H_DIM + h] = a * invL;
  }
  for (int s = tid; s < S_DIM; s += 128) {
    const float sc = weights[(size_t)b * S_DIM + s];
    weights[(size_t)b * S_DIM + s] = __expf(sc - mg) * invL;
  }
}

extern "C" void kernel_launch(void* const* d_in, const int* in_sizes, int n_in,
                              void* d_out, int out_size, void* d_ws, size_t ws_size,
                              hipStream_t stream) {
  const float* x = (const float*)d_in[0];   // [64,4096,512] fp32
  const float* w = (const float*)d_in[1];   // [512] fp32
  float* out     = (float*)d_out;
  float* pooled  = out;                     // [64,512]
  float* weights = out + (size_t)B_DIM * H_DIM;  // [64,4096]

  float* partial = (float*)d_ws;                               // B*CHUNKS*H
  float* ml      = partial + (size_t)B_DIM * CHUNKS * H_DIM;   // B*CHUNKS*2

  attn_pass1<<<dim3(CHUNKS, B_DIM), BLK, 0, stream>>>(x, w, weights, partial, ml);
  attn_pass2<<<B_DIM, 128, 0, stream>>>(pooled, weights, partial, ml);
}